// ResidualGNNs_38680475467998
// MI455X (gfx1250) — compile-verified
//
// ResidualGNN pipeline for MI455X (gfx1250), wave32 + WMMA f32 16x16x4 +
// async global->LDS staging.
//
// Memory-bound analysis (23.3 TB/s HBM): x = 256MB, l1_w = 512MB dominate.
// All GEMM reductions use V_WMMA_F32_16X16X4_F32 (fp32 matrix pipe -> no
// precision loss, no extra traffic). Additive constants feeding a
// BatchNorm-over-batch cancel (uniform column shift), so fc1 skips l1_b /
// bn_b / the -mean*a terms; the multiplicative BN factor a[c] is folded into
// the B (l1_w) operand so the A-tile gather is a pure byte copy and can go
// through GLOBAL_LOAD_ASYNC_TO_LDS_B32 (ASYNCcnt) without touching VGPRs.

#include <hip/hip_runtime.h>
#include <cstdint>
#include <cstddef>

#define G_    64
#define NF_   1000
#define HC_   64
#define HID_  256
#define N_    (G_ * NF_)            // 64000 nodes
#define E_    (N_ * 32)             // 2,048,000 edges
#define TRIU_ 499500
#define EPS_  1e-5f
#define CK_   512                   // fc1 K-chunk per block (LDS = 512*65*4 = 133KB, OK on 320KB WGP)

typedef __attribute__((ext_vector_type(2))) float v2f;
typedef __attribute__((ext_vector_type(8))) float v8f;

// D(16x16,f32) = A(16x4,f32) * B(4x16,f32) + C.
// Lane layouts per CDNA5 ISA 7.12.2:
//   A: lane l -> M = l&15, VGPR0 = K(kh), VGPR1 = K(kh+1), kh = (l>>4)*2
//   B: lane l -> N = l&15, VGPR0 = K(kh), VGPR1 = K(kh+1)
//   C/D: vgpr v, lane l -> M = v + 8*(l>>4), N = l&15
__device__ __forceinline__ v8f wmma4(v2f a, v2f b, v8f c) {
#if defined(__HIP_DEVICE_COMPILE__) && __has_builtin(__builtin_amdgcn_wmma_f32_16x16x4_f32)
  return __builtin_amdgcn_wmma_f32_16x16x4_f32(false, a, false, b, (short)0, c, false, false);
#else
  // host-pass / probe-mismatch fallback stub (not a correct matrix op)
  c[0] += a[0] * b[0] + a[1] * b[1];
  return c;
#endif
}

// ---------------- utility kernels ----------------

__global__ void k_zero(float* __restrict__ p, int n) {
  int t = blockIdx.x * blockDim.x + threadIdx.x;
  if (t < n) p[t] = 0.0f;
}

__global__ void k_deg_init(float* __restrict__ deg) {
  int t = blockIdx.x * blockDim.x + threadIdx.x;
  if (t < N_) deg[t] = 1.0f;   // self-loop
}

__global__ void k_deg_count(const int* __restrict__ dst, float* __restrict__ deg) {
  int t = blockIdx.x * blockDim.x + threadIdx.x;
  if (t < E_) atomicAdd(&deg[dst[t]], 1.0f);
}

__global__ void k_dinv(float* __restrict__ deg) {
  int t = blockIdx.x * blockDim.x + threadIdx.x;
  if (t < N_) deg[t] = rsqrtf(deg[t]);   // deg >= 1 always
}

// ---------------- GCN: H = Xin @ W  (WMMA f32) ----------------
// grid.x = N_/32 blocks, 256 threads = 8 waves; wave w: mtile = 2*bid + (w>>2), ntile = w&3.
__global__ void k_gcn_gemm(const float* __restrict__ X, int ld, int K,
                           const float* __restrict__ W,   // K x 64 row-major
                           float* __restrict__ H) {       // N x 64
  int lane = threadIdx.x & 31, wave = threadIdx.x >> 5;
  int m0 = (blockIdx.x * 2 + (wave >> 2)) * 16;
  int n0 = (wave & 3) * 16;
  int mlo = lane & 15;
  int kh  = (lane >> 4) * 2;
  int ncol = n0 + mlo;
  const float* xrow = X + (size_t)(m0 + mlo) * ld;
  v8f acc = {};
  for (int k = 0; k < K; k += 4) {
    v2f a, b;
    a[0] = xrow[k + kh];
    a[1] = xrow[k + kh + 1];
    b[0] = W[(size_t)(k + kh) * HC_ + ncol];
    b[1] = W[(size_t)(k + kh + 1) * HC_ + ncol];
    acc = wmma4(a, b, acc);
  }
  int rbase = m0 + ((lane >> 4) << 3);
#pragma unroll
  for (int v = 0; v < 8; ++v)
    H[(size_t)(rbase + v) * HC_ + ncol] = acc[v];
}

// agg[n][c] = h[n][c] * dinv[n]^2   (self-loop term, also initializes agg)
__global__ void k_self_init(const float* __restrict__ h, const float* __restrict__ dinv,
                            float* __restrict__ agg) {
  int t = blockIdx.x * blockDim.x + threadIdx.x;
  if (t >= N_ * HC_) return;
  int n = t >> 6;
  float w = dinv[n];
  agg[t] = h[t] * w * w;
}

// per (edge, 4-channel group): agg[dst] += h[src] * dinv[src]*dinv[dst]
__global__ void k_edge_scatter(const int* __restrict__ src, const int* __restrict__ dst,
                               const float* __restrict__ h, const float* __restrict__ dinv,
                               float* __restrict__ agg) {
  long long t = (long long)blockIdx.x * blockDim.x + threadIdx.x;
  if (t >= (long long)E_ * 16) return;
  int e = (int)(t >> 4), q = (int)(t & 15);
  int s = src[e], d = dst[e];
  float w = dinv[s] * dinv[d];
  const float4* hp = reinterpret_cast<const float4*>(h + (size_t)s * HC_);
  float4 v = hp[q];
  float* ap = agg + (size_t)d * HC_ + q * 4;
  atomicAdd(ap + 0, v.x * w);
  atomicAdd(ap + 1, v.y * w);
  atomicAdd(ap + 2, v.z * w);
  atomicAdd(ap + 3, v.w * w);
}

// xs = tanh(agg + b); pool[batch][loff+c] += xs / NF
__global__ void k_finish(const float* __restrict__ agg, const float* __restrict__ bias,
                         const int* __restrict__ batch, int loff,
                         float* __restrict__ xs, float* __restrict__ pool) {
  int t = blockIdx.x * blockDim.x + threadIdx.x;
  if (t >= N_ * HC_) return;
  int n = t >> 6, c = t & 63;
  float v = tanhf(agg[t] + bias[c]);
  xs[t] = v;
  atomicAdd(&pool[(size_t)batch[n] * 192 + loff + c], v * (1.0f / NF_));
}

// ---------------- BN over batch axis (64 rows) ----------------
__global__ void k_bn_cols(const float* __restrict__ in, const float* __restrict__ g,
                          const float* __restrict__ b, float* __restrict__ out,
                          int C, int relu) {
  int c = blockIdx.x * blockDim.x + threadIdx.x;
  if (c >= C) return;
  float s = 0.f, sq = 0.f;
  for (int r = 0; r < G_; ++r) { float v = in[(size_t)r * C + c]; s += v; sq += v * v; }
  float m = s * (1.0f / G_);
  float var = sq * (1.0f / G_) - m * m;
  float a = rsqrtf(var + EPS_) * g[c];
  float d = b[c] - m * a;
  for (int r = 0; r < G_; ++r) {
    float v = in[(size_t)r * C + c] * a + d;
    out[(size_t)r * C + c] = (relu && v < 0.f) ? 0.f : v;
  }
}

// triu column stats: a[c] = rsqrt(var+eps)*bn_g[c]; colmap[c] = i*NF+j
// (additive BN term cancels through the post-fc1 BatchNorm -> skipped)
__global__ void k_triu_stats(const float* __restrict__ x, const float* __restrict__ bng,
                             float* __restrict__ atri, int* __restrict__ colmap) {
  int i = blockIdx.x;                       // 0 .. NF_-2
  int rowbase = i * (NF_ - 1) - (i * (i - 1)) / 2;
  for (int j = i + 1 + threadIdx.x; j < NF_; j += blockDim.x) {
    int c = rowbase + (j - i - 1);
    int off = i * NF_ + j;
    float s = 0.f, sq = 0.f;
    for (int g = 0; g < G_; ++g) {
      float v = x[(size_t)g * (NF_ * NF_) + off];
      s += v; sq += v * v;
    }
    float m = s * (1.0f / G_);
    float var = sq * (1.0f / G_) - m * m;
    atri[c] = rsqrtf(var + EPS_) * bng[c];
    colmap[c] = off;
  }
}

// ---------------- fc1: out1[64x256] += xt[:, c0:c0+CK] @ (a*l1_w)[c0:c0+CK, :] ------------
// Split-K over TRIU columns. The A tile (gathered raw x) is staged into LDS with
// GLOBAL_LOAD_ASYNC_TO_LDS_B32 (per-lane global addr -> per-lane LDS addr, ASYNCcnt);
// the BN scale a[c] rides on the B (l1_w) operand. l1_w is streamed exactly once.
__global__ void k_fc1(const float* __restrict__ x, const int* __restrict__ colmap,
                      const float* __restrict__ atri, const float* __restrict__ l1w,
                      float* __restrict__ out1) {
  extern __shared__ float zn[];             // [CK_][65] padded, 133 KB
  int c0 = blockIdx.x * CK_;
  int lane = threadIdx.x & 31, wave = threadIdx.x >> 5;

  // stage: wave w handles graphs g = w, w+8, ...; lanes sweep columns.
#if defined(__HIP_DEVICE_COMPILE__)
  unsigned ldsbase = (unsigned)__builtin_amdgcn_groupstaticsize();  // dynamic LDS base
  for (int g = wave; g < G_; g += 8) {
    const float* xg = x + (size_t)g * (NF_ * NF_);
    for (int cc = lane; cc < CK_; cc += 32) {
      int c = c0 + cc;
      if (c < TRIU_) {
        unsigned dsoff = ldsbase + (unsigned)(cc * 65 + g) * 4u;
        const float* gp = xg + colmap[c];
        // async copy global -> LDS, no VGPR round-trip (tracked on ASYNCcnt)
        asm volatile("global_load_async_to_lds_b32 %0, %1, off"
                     :: "v"(dsoff), "v"(gp)
                     : "memory");
      } else {
        zn[cc * 65 + g] = 0.f;              // tail of last block (DScnt, compiler-tracked)
      }
    }
  }
  asm volatile("s_wait_asynccnt 0" ::: "memory");
#else
  for (int g = wave; g < G_; g += 8) {
    const float* xg = x + (size_t)g * (NF_ * NF_);
    for (int cc = lane; cc < CK_; cc += 32) {
      int c = c0 + cc;
      zn[cc * 65 + g] = (c < TRIU_) ? xg[colmap[c]] : 0.f;
    }
  }
#endif
  __syncthreads();

  // wave w owns output columns n = [32w, 32w+32): two 16-wide ntiles, all 4 mtiles (graphs)
  v8f acc[4][2] = {};
  int kh = (lane >> 4) * 2;
  int nlo = lane & 15;
  for (int kk = 0; kk < CK_; kk += 4) {
    int r0 = c0 + kk + kh;
    v2f a[4];
#pragma unroll
    for (int mt = 0; mt < 4; ++mt) {
      a[mt][0] = zn[(kk + kh) * 65 + mt * 16 + nlo];
      a[mt][1] = zn[(kk + kh + 1) * 65 + mt * 16 + nlo];
    }
    float s0 = (r0 < TRIU_)     ? atri[r0]     : 0.f;   // BN scale folded into B
    float s1 = (r0 + 1 < TRIU_) ? atri[r0 + 1] : 0.f;
    if ((kk & 31) == 0 && r0 + 256 < TRIU_)
      __builtin_prefetch(l1w + (size_t)(r0 + 256) * HID_ + wave * 32, 0, 0);
#pragma unroll
    for (int nt = 0; nt < 2; ++nt) {
      int n0 = (wave * 2 + nt) * 16;
      v2f b;
      b[0] = (r0 < TRIU_)     ? s0 * l1w[(size_t)r0 * HID_ + n0 + nlo]       : 0.f;
      b[1] = (r0 + 1 < TRIU_) ? s1 * l1w[(size_t)(r0 + 1) * HID_ + n0 + nlo] : 0.f;
#pragma unroll
      for (int mt = 0; mt < 4; ++mt)
        acc[mt][nt] = wmma4(a[mt], b, acc[mt][nt]);
    }
  }

  // split-K reduction into the 64x256 accumulator (tiny, L2-resident)
#pragma unroll
  for (int mt = 0; mt < 4; ++mt)
#pragma unroll
    for (int nt = 0; nt < 2; ++nt) {
      int n = (wave * 2 + nt) * 16 + nlo;
      int rb = mt * 16 + ((lane >> 4) << 3);
#pragma unroll
      for (int v = 0; v < 8; ++v)
        atomicAdd(&out1[(size_t)(rb + v) * HID_ + n], acc[mt][nt][v]);
    }
}

// fc1 contribution of the pooled-h block: out1 += znh[64x192] @ l1_w[TRIU:,:]
__global__ void k_fc1_hpart(const float* __restrict__ znh, const float* __restrict__ l1w,
                            float* __restrict__ out1) {
  int t = blockIdx.x * blockDim.x + threadIdx.x;
  if (t >= G_ * HID_) return;
  int g = t >> 8, o = t & 255;
  float s = 0.f;
  for (int c = 0; c < 192; ++c)
    s += znh[g * 192 + c] * l1w[(size_t)(TRIU_ + c) * HID_ + o];
  out1[t] += s;
}

// small dense layer: out[64 x Cout] = in[64 x Cin] @ W (bias cancels in next BN)
__global__ void k_small_gemm(const float* __restrict__ in, const float* __restrict__ W,
                             float* __restrict__ out, int Cin, int Cout) {
  int t = blockIdx.x * blockDim.x + threadIdx.x;
  if (t >= G_ * Cout) return;
  int g = t / Cout, o = t % Cout;
  float s = 0.f;
  for (int c = 0; c < Cin; ++c) s += in[g * Cin + c] * W[(size_t)c * Cout + o];
  out[t] = s;
}

// final layer with bias (no BN after -> bias kept)
__global__ void k_final(const float* __restrict__ z, const float* __restrict__ W,
                        const float* __restrict__ b, float* __restrict__ out) {
  int t = threadIdx.x;
  if (t >= G_ * 2) return;
  int g = t >> 1, o = t & 1;
  float s = b[o];
  for (int c = 0; c < 128; ++c) s += z[g * 128 + c] * W[c * 2 + o];
  out[t] = s;
}

// ---------------- launch ----------------
extern "C" void kernel_launch(void* const* d_in, const int* in_sizes, int n_in,
                              void* d_out, int out_size, void* d_ws, size_t ws_size,
                              hipStream_t stream) {
  (void)in_sizes; (void)n_in; (void)out_size; (void)ws_size;
  const float* x     = (const float*)d_in[0];
  const int*   ei    = (const int*)d_in[1];
  const int*   batch = (const int*)d_in[2];
  const float* W[3]  = {(const float*)d_in[3], (const float*)d_in[5], (const float*)d_in[7]};
  const float* bb[3] = {(const float*)d_in[4], (const float*)d_in[6], (const float*)d_in[8]};
  const float* bn_g  = (const float*)d_in[9];    // bn_b (d_in[10]) cancels in BN1
  const float* bnh_g = (const float*)d_in[11];
  const float* bnh_b = (const float*)d_in[12];
  const float* l1w   = (const float*)d_in[13];   // l1_b (14) cancels in BN1
  const float* bm1_g = (const float*)d_in[15];
  const float* bm1_b = (const float*)d_in[16];
  const float* l2w   = (const float*)d_in[17];   // l2_b cancels in BN2
  const float* bm2_g = (const float*)d_in[19];
  const float* bm2_b = (const float*)d_in[20];
  const float* l3w   = (const float*)d_in[21];   // l3_b cancels in BN3
  const float* bm3_g = (const float*)d_in[23];
  const float* bm3_b = (const float*)d_in[24];
  const float* l4w   = (const float*)d_in[25];
  const float* l4b   = (const float*)d_in[26];
  float* out = (float*)d_out;

  float* f = (float*)d_ws;
  size_t off = 0;
  auto alloc = [&](size_t n) { float* p = f + off; off += (n + 63) & ~size_t(63); return p; };
  float* hraw   = alloc((size_t)N_ * HC_);
  float* agg    = alloc((size_t)N_ * HC_);
  float* xsbuf0 = alloc((size_t)N_ * HC_);
  float* xsbuf1 = alloc((size_t)N_ * HC_);
  float* dinv   = alloc(N_);
  float* pool   = alloc(G_ * 192);
  float* znh    = alloc(G_ * 192);
  float* atri   = alloc(TRIU_);
  int*   colmap = (int*)alloc(TRIU_);
  float* out1   = alloc(G_ * HID_);
  float* z1n    = alloc(G_ * HID_);
  float* z2     = alloc(G_ * 128);
  float* z2n    = alloc(G_ * 128);
  float* z3     = alloc(G_ * 128);
  float* z3n    = alloc(G_ * 128);

  const int* src = ei;
  const int* dst = ei + E_;

  // degree -> dinv (shared by all 3 GCN layers)
  k_deg_init<<<(N_ + 255) / 256, 256, 0, stream>>>(dinv);
  k_deg_count<<<(E_ + 255) / 256, 256, 0, stream>>>(dst, dinv);
  k_dinv<<<(N_ + 255) / 256, 256, 0, stream>>>(dinv);
  k_zero<<<(G_ * 192 + 255) / 256, 256, 0, stream>>>(pool, G_ * 192);

  // 3 GCN layers (layer0 consumes x [ld=K=1000], later layers ld=K=64)
  const float* xin = x; int ldk = NF_;
  float* xsout[3] = {xsbuf0, xsbuf1, xsbuf0};
  for (int l = 0; l < 3; ++l) {
    k_gcn_gemm<<<N_ / 32, 256, 0, stream>>>(xin, ldk, ldk, W[l], hraw);
    k_self_init<<<(N_ * HC_ + 255) / 256, 256, 0, stream>>>(hraw, dinv, agg);
    k_edge_scatter<<<(int)(((long long)E_ * 16 + 255) / 256), 256, 0, stream>>>(
        src, dst, hraw, dinv, agg);
    k_finish<<<(N_ * HC_ + 255) / 256, 256, 0, stream>>>(agg, bb[l], batch, l * HC_,
                                                         xsout[l], pool);
    xin = xsout[l]; ldk = HC_;
  }

  // head
  k_bn_cols<<<(192 + 63) / 64, 64, 0, stream>>>(pool, bnh_g, bnh_b, znh, 192, 0);
  k_triu_stats<<<NF_ - 1, 256, 0, stream>>>(x, bn_g, atri, colmap);
  k_zero<<<(G_ * HID_ + 255) / 256, 256, 0, stream>>>(out1, G_ * HID_);
  k_fc1<<<(TRIU_ + CK_ - 1) / CK_, 256, CK_ * 65 * sizeof(float), stream>>>(
      x, colmap, atri, l1w, out1);
  k_fc1_hpart<<<(G_ * HID_ + 255) / 256, 256, 0, stream>>>(znh, l1w, out1);
  k_bn_cols<<<(HID_ + 63) / 64, 64, 0, stream>>>(out1, bm1_g, bm1_b, z1n, HID_, 1);
  k_small_gemm<<<(G_ * 128 + 255) / 256, 256, 0, stream>>>(z1n, l2w, z2, 256, 128);
  k_bn_cols<<<2, 64, 0, stream>>>(z2, bm2_g, bm2_b, z2n, 128, 1);
  k_small_gemm<<<(G_ * 128 + 255) / 256, 256, 0, stream>>>(z2n, l3w, z3, 128, 128);
  k_bn_cols<<<2, 64, 0, stream>>>(z3, bm3_g, bm3_b, z3n, 128, 1);
  k_final<<<1, 128, 0, stream>>>(z3n, l4w, l4b, out);
}